// DistanceNetwork_31250182046456
// MI455X (gfx1250) — compile-verified
//
#include <hip/hip_runtime.h>
#include <hip/hip_bf16.h>
#include <stdint.h>

typedef __attribute__((ext_vector_type(2))) float v2f;
typedef __attribute__((ext_vector_type(4))) float f4;
typedef __attribute__((ext_vector_type(8))) float v8f;

#define N_ROWS 512
#define BATCH  256
#define DIM    1024
#define EPS_F  1e-10f

// Block: 256 threads = 8 waves (wave32). Each wave: 16 rows of n for one b.
// grid = (N/128, B).
__global__ __launch_bounds__(256)
void DistanceNetwork_kernel(const float* __restrict__ S,   // [N, B, D]
                            const float* __restrict__ T,   // [B, D]
                            float* __restrict__ out)       // [B, N]
{
    __shared__ float sT[DIM];          // targets row for this b (4 KB)
    __shared__ float sRed[8][32];      // per-wave: 16 dot + 16 sumsq

    const int b    = blockIdx.y;
    const int t    = threadIdx.x;
    const int wave = t >> 5;
    const int lane = t & 31;

    // ---- Stage T[b, :] into LDS via CDNA5 async global->LDS copy ----
    {
        const f4* src = (const f4*)(T + (size_t)b * DIM) + t;   // one float4/thread
        uint32_t ldsDst = (uint32_t)(uintptr_t)(&sT[t * 4]);    // low 32 bits = LDS offset
        uint64_t gaddr  = (uint64_t)(uintptr_t)src;
        asm volatile("global_load_async_to_lds_b128 %0, %1, off"
                     :: "v"(ldsDst), "v"(gaddr) : "memory");
        asm volatile("s_wait_asynccnt 0x0" ::: "memory");
    }
    __syncthreads();

    // ---- Streaming accumulation ----
    const int n0 = blockIdx.x * 128 + wave * 16;
    const int m  = lane & 15;          // row within the 16-row tile
    const int h  = lane >> 4;          // lane half: covers D-quarters 2h and 2h+1

    const float* row = S + ((size_t)(n0 + m) * BATCH + b) * DIM;
    const f4* r0 = (const f4*)(row + (2 * h)     * 256);
    const f4* r1 = (const f4*)(row + (2 * h + 1) * 256);
    const f4* t0 = (const f4*)(sT  + (2 * h)     * 256);
    const f4* t1 = (const f4*)(sT  + (2 * h + 1) * 256);

    float pd0 = 0.f, pd1 = 0.f, pq0 = 0.f, pq1 = 0.f;
    #pragma unroll 8
    for (int i = 0; i < 64; ++i) {
        f4 a = __builtin_nontemporal_load(&r0[i]);   // streaming: NT hint
        f4 c = t0[i];                                // LDS broadcast within half
        pd0 += a.x * c.x + a.y * c.y + a.z * c.z + a.w * c.w;
        pq0 += a.x * a.x + a.y * a.y + a.z * a.z + a.w * a.w;
        f4 a2 = __builtin_nontemporal_load(&r1[i]);
        f4 c2 = t1[i];
        pd1 += a2.x * c2.x + a2.y * c2.y + a2.z * c2.z + a2.w * c2.w;
        pq1 += a2.x * a2.x + a2.y * a2.y + a2.z * a2.z + a2.w * a2.w;
    }

    // ---- WMMA cross-lane reduction: D = A x ones ----
    // A layout (16x4 f32): lanes 0-15 hold M=lane, VGPR0=K0, VGPR1=K1;
    // lanes 16-31 hold M=lane-16, VGPR0=K2, VGPR1=K3. Our (pd0,pd1) = quarters
    // (2h, 2h+1) matches exactly. All-ones B is layout-independent.
    v2f Ad = { pd0, pd1 };
    v2f Aq = { pq0, pq1 };
    v2f Bone = { 1.0f, 1.0f };
    v8f Cd = {};
    v8f Cq = {};
    Cd = __builtin_amdgcn_wmma_f32_16x16x4_f32(false, Ad, false, Bone,
                                               (short)0, Cd, false, false);
    Cq = __builtin_amdgcn_wmma_f32_16x16x4_f32(false, Aq, false, Bone,
                                               (short)0, Cq, false, false);

    // D layout: VGPR r = row r (lanes 0-15) / row r+8 (lanes 16-31); every
    // column identical. Lane 0 -> rows 0-7, lane 16 -> rows 8-15.
    float* red = sRed[wave];
    if ((lane & 15) == 0) {
        #pragma unroll
        for (int r = 0; r < 8; ++r) {
            red[h * 8 + r]      = Cd[r];
            red[16 + h * 8 + r] = Cq[r];
        }
    }
    __builtin_amdgcn_wave_barrier();

    if (lane < 16) {
        float dot = red[lane];
        float sq  = red[16 + lane];
        out[(size_t)b * N_ROWS + n0 + lane] = dot * __frsqrt_rn(fmaxf(sq, EPS_F));
    }
}

extern "C" void kernel_launch(void* const* d_in, const int* in_sizes, int n_in,
                              void* d_out, int out_size, void* d_ws, size_t ws_size,
                              hipStream_t stream) {
    const float* S = (const float*)d_in[0];   // [512, 256, 1024]
    const float* T = (const float*)d_in[1];   // [256, 1024]
    float* out = (float*)d_out;               // [256, 512]
    dim3 grid(N_ROWS / 128, BATCH);           // (4, 256)
    DistanceNetwork_kernel<<<grid, 256, 0, stream>>>(S, T, out);
}